// VolumeDecoder_9113920602184
// MI455X (gfx1250) — compile-verified
//
#include <hip/hip_runtime.h>
#include <hip/hip_bf16.h>

// ---------------------------------------------------------------------------
// VolumeDecoder: multi-cam voxel lift-splat.
//   S=6 cams, C=64 ch, feat 44x80; voxel grid Z=16,Y=100,X=100 (160000 voxels)
//   out (1, C, X, Y, Z) float32, 10,240,000 elements.
// Projection (4x4 matrix x homogeneous voxel center) is done with
// V_WMMA_F32_16X16X4_F32; bilinear gather + camera-average done per-lane.
// ---------------------------------------------------------------------------

#define S_CAMS 6
#define C_CH   64
#define HF     44
#define WF     80
#define ZD     16
#define YD     100
#define XD     100
#define NVOX   (ZD * YD * XD)          // 160000
#define CH_STRIDE (HF * WF)            // 3520
#define CAM_STRIDE (C_CH * HF * WF)    // 225280

typedef __attribute__((ext_vector_type(2))) float v2f;
typedef __attribute__((ext_vector_type(8))) float v8f;

// ---------------------------------------------------------------------------
// Setup: build 32 A-matrix rows (24 used, 8 zero pad) into d_ws:
//   rows 4s+0..4s+2 : rows 0..2 of  P_s = diag(.125,.125,1,1)*K_s * inv(T_s)
//   row  4s+3       : row 2 of inv(T_s)   (z_cam row)
// ---------------------------------------------------------------------------
__global__ void vd_setup_mats(const float* __restrict__ pix,
                              const float* __restrict__ c0X,
                              float* __restrict__ Atab) {
    if (threadIdx.x != 0 || blockIdx.x != 0) return;
    for (int i = 0; i < 32 * 4; ++i) Atab[i] = 0.0f;
    for (int s = 0; s < S_CAMS; ++s) {
        const float* T = c0X + s * 16;
        const float* K = pix + s * 16;
        float inv[4][4];
        for (int i = 0; i < 4; ++i)
            for (int j = 0; j < 4; ++j) inv[i][j] = 0.0f;
        for (int i = 0; i < 3; ++i)
            for (int j = 0; j < 3; ++j) inv[i][j] = T[j * 4 + i];     // R^T
        for (int i = 0; i < 3; ++i)
            inv[i][3] = -(inv[i][0] * T[0 * 4 + 3] +
                          inv[i][1] * T[1 * 4 + 3] +
                          inv[i][2] * T[2 * 4 + 3]);                  // -R^T t
        inv[3][3] = 1.0f;
        float fp[4][4];
        for (int j = 0; j < 4; ++j) {
            fp[0][j] = 0.125f * K[0 * 4 + j];   // sx = 80/640
            fp[1][j] = 0.125f * K[1 * 4 + j];   // sy = 44/352
            fp[2][j] = K[2 * 4 + j];
            fp[3][j] = K[3 * 4 + j];
        }
        for (int i = 0; i < 3; ++i)
            for (int j = 0; j < 4; ++j) {
                float a = 0.0f;
                for (int k = 0; k < 4; ++k) a += fp[i][k] * inv[k][j];
                Atab[(4 * s + i) * 4 + j] = a;
            }
        for (int j = 0; j < 4; ++j)
            Atab[(4 * s + 3) * 4 + j] = inv[2][j];
    }
}

// ---------------------------------------------------------------------------
// Main kernel: one thread per voxel. Output index = c*NVOX + g (coalesced).
// ---------------------------------------------------------------------------
__global__ __launch_bounds__(256) void vd_main(const float* __restrict__ feat,
                                               const float* __restrict__ Atab,
                                               float* __restrict__ out) {
    __shared__ float proj[256][25];   // 24 rows used, pad to 25 (gcd(25,64)=1)

    const int t    = threadIdx.x;
    const int g    = blockIdx.x * 256 + t;
    const int lane = t & 31;
    const int m    = lane & 15;
    const bool hi  = lane >= 16;

    // voxel indices: g = (x*YD + y)*ZD + z  -> matches out flat layout
    const int z = g & (ZD - 1);
    const int r = g >> 4;
    const int y = r % YD;
    const int x = r / YD;
    const float xw = -40.0f + 0.8f * ((float)x + 0.5f);
    const float yw = -40.0f + 0.8f * ((float)y + 0.5f);
    const float zw =  -1.0f + 0.4f * ((float)z + 0.5f);

    // ---- Phase 1: projection via V_WMMA_F32_16X16X4_F32 --------------------
    // A 16x4 f32 layout: VGPR0 = K0 (lanes 0-15) | K2 (lanes 16-31); VGPR1 = K1|K3
    v2f a1, a2;
    a1.x = Atab[m * 4 + (hi ? 2 : 0)];
    a1.y = Atab[m * 4 + (hi ? 3 : 1)];
    a2.x = Atab[(16 + m) * 4 + (hi ? 2 : 0)];
    a2.y = Atab[(16 + m) * 4 + (hi ? 3 : 1)];

    const int wavebase = t & ~31;

#pragma unroll
    for (int G = 0; G < 2; ++G) {
        const int src = (G << 4) | m;                 // lane owning voxel N=m
        const float bx = __shfl(xw, src, 32);
        const float by = __shfl(yw, src, 32);
        const float bz = __shfl(zw, src, 32);
        v2f b;
        b.x = hi ? bz : bx;                           // B 4x16: K0|K2 then K1|K3
        b.y = hi ? 1.0f : by;

        v8f cz = {};
        // D = A*B: rows 0..15 = cams 0..3 (4 rows each)
        v8f d1 = __builtin_amdgcn_wmma_f32_16x16x4_f32(
            false, a1, false, b, (short)0, cz, false, false);
        // rows 16..23 = cams 4..5; rows 24..31 are zero-padded A rows
        v8f d2 = __builtin_amdgcn_wmma_f32_16x16x4_f32(
            false, a2, false, b, (short)0, cz, false, false);

        const int vloc = wavebase + src;
        const int rowbase = hi ? 8 : 0;               // lanes 16-31 hold M=8..15
#pragma unroll
        for (int j = 0; j < 8; ++j) proj[vloc][rowbase + j] = d1[j];
        if (!hi) {
#pragma unroll
            for (int j = 0; j < 8; ++j) proj[vloc][16 + j] = d2[j];
        }
    }
    __syncthreads();

    // ---- Phase 2: bilinear sampling params per camera ----------------------
    float w[S_CAMS][4];
    int   p[S_CAMS][4];
    bool  act[S_CAMS];

#pragma unroll
    for (int s = 0; s < S_CAMS; ++s) {
        const float px = proj[t][4 * s + 0];
        const float py = proj[t][4 * s + 1];
        const float pz = proj[t][4 * s + 2];
        const float zc = proj[t][4 * s + 3];
        const float den = fmaxf(pz, 1e-6f);
        const float xp = px / den;
        const float yp = py / den;
        const bool valid = (xp > -0.5f) && (xp < (float)WF - 0.5f) &&
                           (yp > -0.5f) && (yp < (float)HF - 0.5f) && (zc > 0.0f);
        const float xs = xp - 0.5f, ys = yp - 0.5f;
        const float x0f = floorf(xs), y0f = floorf(ys);
        const float fx = xs - x0f,   fy = ys - y0f;
        const int x0 = (int)x0f, y0 = (int)y0f;
        const int x1 = x0 + 1,   y1 = y0 + 1;
        const bool vx0 = (x0 >= 0) && (x0 < WF);
        const bool vx1 = (x1 >= 0) && (x1 < WF);
        const bool vy0 = (y0 >= 0) && (y0 < HF);
        const bool vy1 = (y1 >= 0) && (y1 < HF);
        const float V = valid ? 1.0f : 0.0f;
        w[s][0] = (1.0f - fx) * (1.0f - fy) * ((vx0 && vy0) ? V : 0.0f);
        w[s][1] = fx * (1.0f - fy)          * ((vx1 && vy0) ? V : 0.0f);
        w[s][2] = (1.0f - fx) * fy          * ((vx0 && vy1) ? V : 0.0f);
        w[s][3] = fx * fy                   * ((vx1 && vy1) ? V : 0.0f);
        const int x0c = min(max(x0, 0), WF - 1);
        const int x1c = min(max(x1, 0), WF - 1);
        const int y0c = min(max(y0, 0), HF - 1);
        const int y1c = min(max(y1, 0), HF - 1);
        const int base = s * CAM_STRIDE;
        p[s][0] = base + y0c * WF + x0c;
        p[s][1] = base + y0c * WF + x1c;
        p[s][2] = base + y1c * WF + x0c;
        p[s][3] = base + y1c * WF + x1c;
        act[s] = valid;
        if (valid) __builtin_prefetch(feat + p[s][0], 0, 3);
    }

    // ---- Channel loop: chunks of 8, cameras averaged with nonzero mask -----
    for (int chunk = 0; chunk < C_CH / 8; ++chunk) {
        float acc[8], cnt[8];
#pragma unroll
        for (int k = 0; k < 8; ++k) { acc[k] = 0.0f; cnt[k] = 0.0f; }
#pragma unroll
        for (int s = 0; s < S_CAMS; ++s) {
            if (!act[s]) continue;
#pragma unroll
            for (int k = 0; k < 8; ++k) {
                const int cbase = (chunk * 8 + k) * CH_STRIDE;
                const float f00 = feat[p[s][0] + cbase];
                const float f10 = feat[p[s][1] + cbase];
                const float f01 = feat[p[s][2] + cbase];
                const float f11 = feat[p[s][3] + cbase];
                const float v = w[s][0] * f00 + w[s][1] * f10 +
                                w[s][2] * f01 + w[s][3] * f11;
                acc[k] += v;
                cnt[k] += (fabsf(v) > 0.0f) ? 1.0f : 0.0f;
            }
        }
#pragma unroll
        for (int k = 0; k < 8; ++k)
            out[(chunk * 8 + k) * NVOX + g] = acc[k] / (cnt[k] + 1e-6f);
    }
}

// ---------------------------------------------------------------------------
extern "C" void kernel_launch(void* const* d_in, const int* in_sizes, int n_in,
                              void* d_out, int out_size, void* d_ws, size_t ws_size,
                              hipStream_t stream) {
    (void)in_sizes; (void)n_in; (void)out_size; (void)ws_size;
    const float* feat = (const float*)d_in[0];   // (S, C, HF, WF)
    const float* pix  = (const float*)d_in[1];   // (S, 4, 4)
    const float* c0X  = (const float*)d_in[2];   // (S, 4, 4)
    float* out  = (float*)d_out;                 // (1, C, X, Y, Z)
    float* Atab = (float*)d_ws;                  // 32 rows x 4 floats

    vd_setup_mats<<<1, 32, 0, stream>>>(pix, c0X, Atab);
    vd_main<<<NVOX / 256, 256, 0, stream>>>(feat, Atab, out);
}